// KernelNN_35347580846814
// MI455X (gfx1250) — compile-verified
//
#include <hip/hip_runtime.h>

#define N_NODES 20000
#define N_EDGES 100000
#define WIDTH 64
#define KER_IN 6
#define KER_HID 16
#define DEPTH 6
#define GCOLS 1152   // 16*64 (k-slices) + 64 (k2_b image) + 64 (root_w image)
#define NT 10        // node tiles per GEMM block (1250 tiles / 10 = 125 blocks)

typedef __attribute__((ext_vector_type(2))) float v2f;
typedef __attribute__((ext_vector_type(8))) float v8f;

// -------------------------------------------------------------------------
// G[n, c] = sum_i h[n,i] * M[i,c]  where
//   c in [0,1024):    M[i,c] = k2_w[k*4096 + i*64 + o],  k=c>>6, o=c&63
//   c in [1024,1088): M[i,c] = k2_b[i*64 + (c-1024)]
//   c in [1088,1152): M[i,c] = root_w[i*64 + (c-1088)]
// One block = 4 waves; each wave owns a 16-wide column tile and sweeps NT
// node tiles. B fragments register-resident (loaded once); A double-buffered
// in LDS so the prefetch of tile nt+1 overlaps the WMMA chain of tile nt.
// K=64 swept with 16x V_WMMA_F32_16X16X4_F32 (f32 precision preserved).
// -------------------------------------------------------------------------
__global__ __launch_bounds__(128)
void gemm_G_kernel(const float* __restrict__ h,
                   const float* __restrict__ k2w,
                   const float* __restrict__ k2b,
                   const float* __restrict__ rootw,
                   float* __restrict__ G)
{
    __shared__ float lA[2][16][WIDTH];         // 2 x (16 nodes x 64 feats) = 8 KB

    const int wave = threadIdx.x >> 5;         // 0..3
    const int lane = threadIdx.x & 31;

    const int ct = blockIdx.y * 4 + wave;      // column tile 0..71
    const int n  = lane & 15;                  // C column within tile
    const int m  = lane & 15;                  // A row (M) per ISA layout
    const int koff = (lane >> 4) << 1;         // upper half-wave holds K+2,K+3
    const int c  = ct * 16 + n;                // global column

    // region select: uniform per wave (tiles never straddle 1024/1088)
    const float* bptr;
    int bbase;
    if (c < 1024)       { bptr = k2w;   bbase = (c >> 6) * 4096 + (c & 63); }
    else if (c < 1088)  { bptr = k2b;   bbase = c - 1024; }
    else                { bptr = rootw; bbase = c - 1088; }

    // B fragments for all 16 K-steps, register-resident, reused NT times
    v2f bfrag[16];
#pragma unroll
    for (int s = 0; s < 16; ++s) {
        bfrag[s].x = bptr[bbase + (s * 4 + koff) * 64];
        bfrag[s].y = bptr[bbase + (s * 4 + koff + 1) * 64];
    }

    const int tile0 = blockIdx.x * NT;

    // stage first A tile
    for (int idx = threadIdx.x; idx < 16 * WIDTH; idx += 128) {
        lA[0][idx >> 6][idx & 63] =
            h[(tile0 * 16 + (idx >> 6)) * WIDTH + (idx & 63)];
    }
    __syncthreads();

    const int rowBase = (lane >> 4) * 8;
    for (int nt = 0; nt < NT; ++nt) {
        const int buf = nt & 1;
        // prefetch next A tile into the other buffer (overlaps WMMA below)
        if (nt + 1 < NT) {
            for (int idx = threadIdx.x; idx < 16 * WIDTH; idx += 128) {
                lA[buf ^ 1][idx >> 6][idx & 63] =
                    h[((tile0 + nt + 1) * 16 + (idx >> 6)) * WIDTH + (idx & 63)];
            }
        }

        v8f acc = {};
#pragma unroll
        for (int s = 0; s < 16; ++s) {
            v2f a;
            a.x = lA[buf][m][s * 4 + koff];
            a.y = lA[buf][m][s * 4 + koff + 1];
            acc = __builtin_amdgcn_wmma_f32_16x16x4_f32(
                      false, a, false, bfrag[s], (short)0, acc, false, false);
        }

        // C/D layout: VGPR r holds row r (lanes 0-15) / row 8+r (lanes 16-31)
        const int nodeBase = (tile0 + nt) * 16;
#pragma unroll
        for (int r = 0; r < 8; ++r) {
            G[(size_t)(nodeBase + rowBase + r) * GCOLS + ct * 16 + n] = acc[r];
        }
        __syncthreads();   // prefetch complete + safe to overwrite read buffer
    }
}

// -------------------------------------------------------------------------
// per-edge: msg_e[o] = G[src,1024+o] + sum_k ek[e,k] * G[src, k*64+o]
// then segment-sum via float atomics into agg[dst]. 64 threads per edge.
// G (92 MB) is L2-resident, so the gather is L2-served.
// -------------------------------------------------------------------------
__global__ __launch_bounds__(256)
void edge_msg_kernel(const float* __restrict__ G,
                     const float* __restrict__ ek,
                     const int* __restrict__ src,
                     const int* __restrict__ dst,
                     float* __restrict__ agg)
{
    int t = blockIdx.x * blockDim.x + threadIdx.x;
    int e = t >> 6;
    int o = t & 63;
    if (e >= N_EDGES) return;

    const float* g = G + (size_t)src[e] * GCOLS;
    const float4* ekv = (const float4*)(ek + e * KER_HID);
    float4 e0 = ekv[0], e1 = ekv[1], e2 = ekv[2], e3 = ekv[3];

    float msg = g[1024 + o];                   // h_src @ reshape(k2_b)
    msg += e0.x * g[0 * 64 + o];  msg += e0.y * g[1 * 64 + o];
    msg += e0.z * g[2 * 64 + o];  msg += e0.w * g[3 * 64 + o];
    msg += e1.x * g[4 * 64 + o];  msg += e1.y * g[5 * 64 + o];
    msg += e1.z * g[6 * 64 + o];  msg += e1.w * g[7 * 64 + o];
    msg += e2.x * g[8 * 64 + o];  msg += e2.y * g[9 * 64 + o];
    msg += e2.z * g[10 * 64 + o]; msg += e2.w * g[11 * 64 + o];
    msg += e3.x * g[12 * 64 + o]; msg += e3.y * g[13 * 64 + o];
    msg += e3.z * g[14 * 64 + o]; msg += e3.w * g[15 * 64 + o];

    atomicAdd(&agg[dst[e] * WIDTH + o], msg);
}

// -------------------------------------------------------------------------
// h_next = relu(agg/denom + (h @ root_w) + conv_b); root term lives in G.
// -------------------------------------------------------------------------
__global__ void node_update_kernel(const float* __restrict__ agg,
                                   const float* __restrict__ G,
                                   const float* __restrict__ denom,
                                   const float* __restrict__ conv_b,
                                   float* __restrict__ h_next)
{
    int t = blockIdx.x * blockDim.x + threadIdx.x;
    if (t >= N_NODES * WIDTH) return;
    int n = t >> 6, o = t & 63;
    float v = agg[t] / denom[n] + G[(size_t)n * GCOLS + 1088 + o] + conv_b[o];
    h_next[t] = fmaxf(v, 0.0f);
}

// ------------------------- setup kernels ---------------------------------
__global__ void init_h_kernel(const float* __restrict__ x,
                              const float* __restrict__ fc1w,
                              const float* __restrict__ fc1b,
                              float* __restrict__ h)
{
    int t = blockIdx.x * blockDim.x + threadIdx.x;
    if (t >= N_NODES * WIDTH) return;
    int n = t >> 6, o = t & 63;
    h[t] = x[n] * fc1w[o] + fc1b[o];
}

__global__ void init_ek_kernel(const float* __restrict__ ea,
                               const float* __restrict__ k1w,
                               const float* __restrict__ k1b,
                               float* __restrict__ ek)
{
    int t = blockIdx.x * blockDim.x + threadIdx.x;
    if (t >= N_EDGES * KER_HID) return;
    int e = t >> 4, k = t & 15;
    float s = k1b[k];
#pragma unroll
    for (int i = 0; i < KER_IN; ++i)
        s += ea[e * KER_IN + i] * k1w[i * KER_HID + k];
    ek[t] = fmaxf(s, 0.0f);
}

__global__ void deg_accum_kernel(const int* __restrict__ dst,
                                 float* __restrict__ denom)
{
    int e = blockIdx.x * blockDim.x + threadIdx.x;
    if (e < N_EDGES) atomicAdd(&denom[dst[e]], 1.0f);
}

__global__ void deg_clamp_kernel(float* __restrict__ denom)
{
    int i = blockIdx.x * blockDim.x + threadIdx.x;
    if (i < N_NODES) denom[i] = fmaxf(denom[i], 1.0f);
}

__global__ void final_out_kernel(const float* __restrict__ h,
                                 const float* __restrict__ fc2w,
                                 const float* __restrict__ fc2b,
                                 float* __restrict__ out)
{
    int n = blockIdx.x * blockDim.x + threadIdx.x;
    if (n >= N_NODES) return;
    float s = fc2b[0];
#pragma unroll
    for (int o = 0; o < WIDTH; ++o)
        s += h[n * WIDTH + o] * fc2w[o];
    out[n] = s;
}

// -------------------------------------------------------------------------
extern "C" void kernel_launch(void* const* d_in, const int* in_sizes, int n_in,
                              void* d_out, int out_size, void* d_ws, size_t ws_size,
                              hipStream_t stream)
{
    const float* x     = (const float*)d_in[0];
    const int*   eidx  = (const int*)  d_in[1];
    const float* ea    = (const float*)d_in[2];
    const float* fc1w  = (const float*)d_in[3];
    const float* fc1b  = (const float*)d_in[4];
    const float* k1w   = (const float*)d_in[5];
    const float* k1b   = (const float*)d_in[6];
    const float* k2w   = (const float*)d_in[7];
    const float* k2b   = (const float*)d_in[8];
    const float* rootw = (const float*)d_in[9];
    const float* convb = (const float*)d_in[10];
    const float* fc2w  = (const float*)d_in[11];
    const float* fc2b  = (const float*)d_in[12];

    const int* src = eidx;
    const int* dst = eidx + N_EDGES;

    float* ws = (float*)d_ws;
    size_t off = 0;
    float* G     = ws + off; off += (size_t)N_NODES * GCOLS;   // 92.2 MB
    float* hA    = ws + off; off += (size_t)N_NODES * WIDTH;
    float* hB    = ws + off; off += (size_t)N_NODES * WIDTH;
    float* agg   = ws + off; off += (size_t)N_NODES * WIDTH;
    float* ek    = ws + off; off += (size_t)N_EDGES * KER_HID;
    float* denom = ws + off; off += (size_t)N_NODES;

    // ---- setup (once) ----
    hipMemsetAsync(denom, 0, N_NODES * sizeof(float), stream);
    deg_accum_kernel<<<(N_EDGES + 255) / 256, 256, 0, stream>>>(dst, denom);
    deg_clamp_kernel<<<(N_NODES + 255) / 256, 256, 0, stream>>>(denom);
    init_h_kernel<<<(N_NODES * WIDTH + 255) / 256, 256, 0, stream>>>(x, fc1w, fc1b, hA);
    init_ek_kernel<<<(N_EDGES * KER_HID + 255) / 256, 256, 0, stream>>>(ea, k1w, k1b, ek);

    float* hc = hA;
    float* hn = hB;
    for (int d = 0; d < DEPTH; ++d) {
        // G = hc @ [K2 | k2_b image | root_w]  (WMMA f32 GEMM)
        gemm_G_kernel<<<dim3(N_NODES / 16 / NT, (GCOLS / 16) / 4), 128, 0, stream>>>(
            hc, k2w, k2b, rootw, G);
        hipMemsetAsync(agg, 0, (size_t)N_NODES * WIDTH * sizeof(float), stream);
        edge_msg_kernel<<<(N_EDGES * WIDTH + 255) / 256, 256, 0, stream>>>(
            G, ek, src, dst, agg);
        node_update_kernel<<<(N_NODES * WIDTH + 255) / 256, 256, 0, stream>>>(
            agg, G, denom, convb, hn);
        float* tmp = hc; hc = hn; hn = tmp;
    }

    final_out_kernel<<<(N_NODES + 255) / 256, 256, 0, stream>>>(
        hc, fc2w, fc2b, (float*)d_out);
}